// WSSparseLinear_59837484367883
// MI455X (gfx1250) — compile-verified
//
#include <hip/hip_runtime.h>

typedef float v2f __attribute__((ext_vector_type(2)));
typedef float v8f __attribute__((ext_vector_type(8)));
typedef __attribute__((address_space(3))) float* lds_fp;

constexpr int NDIM = 4096;        // in_features == out_features
constexpr int BM = 128;           // token rows per block
constexpr int BN = 128;           // output cols per block
constexpr int BK = 32;            // K depth per stage
constexpr int XP = 36;            // X tile row stride (words): rows 16B-aligned, frag reads conflict-free
constexpr int BNP = 144;          // B pair-row stride: khalf row offset = 2*BNP = 288 = 32 (mod 64) banks
constexpr int NSTAGE = NDIM / BK; // 128 K stages

// y[t,o] = sum_i x[t,i] * (W[o,i]*M[o,i]) + bias[o]
// f32 WMMA GEMM; mask fused into B staging; async-to-LDS X path; double-buffered pipeline.
__global__ __launch_bounds__(256, 2) void ws_masked_linear_wmma(
    const float* __restrict__ X,    // [T, N]
    const float* __restrict__ W,    // [N, N] (out, in)
    const float* __restrict__ bias, // [N]
    const float* __restrict__ Mk,   // [N, N] (out, in)
    float* __restrict__ Y)          // [T, N]
{
    __shared__ float Xs[2][BM * XP];            // A tiles, [m][k] row-major, ping-pong
    __shared__ float Bs[2][(BK / 2) * BNP * 2]; // B tiles = (W*M)^T, K-pair-major [k/2][n][2]

    const int tid   = threadIdx.x;
    const int lane  = tid & 31;
    const int wave  = tid >> 5;
    const int m0    = (wave & 3) * 32;   // wave sub-tile rows (2 M-tiles)
    const int n0    = (wave >> 2) * 64;  // wave sub-tile cols (4 N-tiles)
    const int lrow  = lane & 15;
    const int khalf = (lane >> 4) << 1;  // 0 (lanes 0-15) or 2 (lanes 16-31)

    const int gm = blockIdx.y * BM;
    const int gn = blockIdx.x * BN;

    v8f acc[2][4] = {};

    // staging map: thread covers 4 float4s; v = tid*4+t, r = v>>3 (tile row), c4 = v&7 (float4 in row)
    const int vb = tid * 4;

    // ---- async X-tile load: memory -> LDS, tracked on ASYNCcnt ----
    auto stage_x_async = [&](int k0, int buf) {
        #pragma unroll
        for (int t = 0; t < 4; ++t) {
            const int v = vb + t, r = v >> 3, c4 = v & 7;
            const float* g = &X[(size_t)(gm + r) * NDIM + k0 + c4 * 4];
            lds_fp l = (lds_fp)&Xs[buf][r * XP + c4 * 4];
            asm volatile("global_load_async_to_lds_b128 %0, %1, off"
                         :: "v"(l), "v"(g) : "memory");
        }
    };
    auto load_wm = [&](int k0, float4* wr, float4* mr) {
        #pragma unroll
        for (int t = 0; t < 4; ++t) {
            const int v = vb + t, r = v >> 3, c4 = v & 7;
            wr[t] = *(const float4*)&W [(size_t)(gn + r) * NDIM + k0 + c4 * 4];
            mr[t] = *(const float4*)&Mk[(size_t)(gn + r) * NDIM + k0 + c4 * 4];
        }
    };
    auto store_b = [&](int buf, const float4* wr, const float4* mr) {
        #pragma unroll
        for (int t = 0; t < 4; ++t) {
            const int v = vb + t, r = v >> 3, c4 = v & 7;
            v2f p0 = { wr[t].x * mr[t].x, wr[t].y * mr[t].y };  // K = c4*4+0, +1
            v2f p1 = { wr[t].z * mr[t].z, wr[t].w * mr[t].w };  // K = c4*4+2, +3
            *(v2f*)&Bs[buf][((c4 * 2 + 0) * BNP + r) * 2] = p0;
            *(v2f*)&Bs[buf][((c4 * 2 + 1) * BNP + r) * 2] = p1;
        }
    };

    // ---- prologue: stage 0 into buffer 0 ----
    stage_x_async(0, 0);
    {
        float4 wr[4], mr[4];
        load_wm(0, wr, mr);
        store_b(0, wr, mr);
    }
    asm volatile("s_wait_asynccnt 0" ::: "memory");
    __syncthreads();

    int buf = 0;
    for (int s = 0; s < NSTAGE; ++s) {
        const int nb = buf ^ 1;
        float4 wr[4], mr[4];
        if (s + 1 < NSTAGE) {
            stage_x_async((s + 1) * BK, nb);   // async copy next A tile
            load_wm((s + 1) * BK, wr, mr);     // next W/M into registers
        }

        // ---- compute current stage: 8 k-steps x 8 WMMAs ----
        #pragma unroll
        for (int kk = 0; kk < BK; kk += 4) {
            const v2f a0 = *(const v2f*)&Xs[buf][(m0 + lrow)      * XP + kk + khalf];
            const v2f a1 = *(const v2f*)&Xs[buf][(m0 + 16 + lrow) * XP + kk + khalf];
            const int kp = (kk + khalf) >> 1;  // lane's K-pair row
            #pragma unroll
            for (int j = 0; j < 4; ++j) {
                const v2f b = *(const v2f*)&Bs[buf][(kp * BNP + n0 + j * 16 + lrow) * 2];
                acc[0][j] = __builtin_amdgcn_wmma_f32_16x16x4_f32(
                    false, a0, false, b, (short)0, acc[0][j], false, false);
                acc[1][j] = __builtin_amdgcn_wmma_f32_16x16x4_f32(
                    false, a1, false, b, (short)0, acc[1][j], false, false);
            }
        }

        if (s + 1 < NSTAGE) store_b(nb, wr, mr);   // mask-fused B staging for next stage
        asm volatile("s_wait_asynccnt 0" ::: "memory");
        __syncthreads();
        buf = nb;
    }

    // ---- epilogue: bias add + store ----
    // C/D layout: VGPR v holds (M = v + 8*(lane>=16), N = lane%16)
    #pragma unroll
    for (int j = 0; j < 4; ++j) {
        const int col = gn + n0 + j * 16 + lrow;
        const float bv = bias[col];
        #pragma unroll
        for (int i = 0; i < 2; ++i) {
            const int rbase = gm + m0 + i * 16 + ((lane >> 4) << 3);
            #pragma unroll
            for (int v = 0; v < 8; ++v) {
                Y[(size_t)(rbase + v) * NDIM + col] = acc[i][j][v] + bv;
            }
        }
    }
}

extern "C" void kernel_launch(void* const* d_in, const int* in_sizes, int n_in,
                              void* d_out, int out_size, void* d_ws, size_t ws_size,
                              hipStream_t stream) {
    const float* x    = (const float*)d_in[0];
    const float* w    = (const float*)d_in[1];
    const float* bias = (const float*)d_in[2];
    const float* mask = (const float*)d_in[3];
    float* y = (float*)d_out;

    const int tokens = in_sizes[0] / NDIM;     // 16384
    dim3 grid(NDIM / BN, tokens / BM);         // (32, 128)
    ws_masked_linear_wmma<<<grid, 256, 0, stream>>>(x, w, bias, mask, y);
}